// LambdaRankLoss_37984690765980
// MI455X (gfx1250) — compile-verified
//
#include <hip/hip_runtime.h>
#include <hip/hip_bf16.h>
#include <cstdint>

// ---------------------------------------------------------------------------
// LambdaRank lambdas, N = 8192 (float32 in/out).
// 64 KB working set -> L2/LDS resident; ~67M pair evals with one v_exp each
// => transcendental/VALU bound. CDNA5 paths: double-buffered async
// global->LDS tiles (ASYNCcnt pipelining, s_wait_asynccnt 1/0), wide
// b64/b128 async + ds loads, wave32 tiling.
// WMMA inapplicable (nonlinear pairwise op, no shared K dimension).
// ---------------------------------------------------------------------------

#define TILE   256
#define NCHUNK 8

#define GLOBAL_AS __attribute__((address_space(1)))
#define LDS_AS    __attribute__((address_space(3)))

typedef int v2i_t __attribute__((ext_vector_type(2)));
typedef int v4i_t __attribute__((ext_vector_type(4)));

#if defined(__HIP_DEVICE_COMPILE__) && defined(__gfx1250__) && \
    __has_builtin(__builtin_amdgcn_global_load_async_to_lds_b32)
#define HAVE_ASYNC_LDS 1
#else
#define HAVE_ASYNC_LDS 0
#endif

#if HAVE_ASYNC_LDS && __has_builtin(__builtin_amdgcn_global_load_async_to_lds_b64)
#define HAVE_ASYNC_B64 1
#else
#define HAVE_ASYNC_B64 0
#endif

#if HAVE_ASYNC_LDS && __has_builtin(__builtin_amdgcn_global_load_async_to_lds_b128)
#define HAVE_ASYNC_B128 1
#else
#define HAVE_ASYNC_B128 0
#endif

// global -> LDS copies; async on gfx1250 (ASYNCcnt), sync fallback otherwise.
__device__ __forceinline__ void lds_cp_f64(float2* dst, const float2* src) {
#if HAVE_ASYNC_B64
  __builtin_amdgcn_global_load_async_to_lds_b64(
      (GLOBAL_AS v2i_t*)(unsigned long long)(uintptr_t)src,
      (LDS_AS    v2i_t*)(unsigned)(uintptr_t)dst, 0, 0);
#else
  *dst = *src;
#endif
}

__device__ __forceinline__ void lds_cp_f128(float4* dst, const float4* src) {
#if HAVE_ASYNC_B128
  __builtin_amdgcn_global_load_async_to_lds_b128(
      (GLOBAL_AS v4i_t*)(unsigned long long)(uintptr_t)src,
      (LDS_AS    v4i_t*)(unsigned)(uintptr_t)dst, 0, 0);
#else
  *dst = *src;
#endif
}

// Wait until <=0 / <=1 async ops outstanding (per-wave ASYNCcnt, in-order).
__device__ __forceinline__ void lds_cp_wait0() {
#if HAVE_ASYNC_LDS
#if __has_builtin(__builtin_amdgcn_s_wait_asynccnt)
  __builtin_amdgcn_s_wait_asynccnt(0);
#else
  asm volatile("s_wait_asynccnt 0" ::: "memory");
#endif
#endif
}

__device__ __forceinline__ void lds_cp_wait1() {
#if HAVE_ASYNC_LDS
#if __has_builtin(__builtin_amdgcn_s_wait_asynccnt)
  __builtin_amdgcn_s_wait_asynccnt(1);
#else
  asm volatile("s_wait_asynccnt 1" ::: "memory");
#endif
#endif
}

// ---------------------------------------------------------------------------
// Kernel 0: pack {t, p} per index for b64 tile loads in the rank sweep.
// ---------------------------------------------------------------------------
__global__ void lr_pack_kernel(const float* __restrict__ t,
                               const float* __restrict__ p,
                               float2* __restrict__ tp, int n) {
  const int i = blockIdx.x * blockDim.x + threadIdx.x;
  if (i < n) tp[i] = make_float2(t[i], p[i]);
}

// ---------------------------------------------------------------------------
// Kernel 1: O(N^2) counting ranks, 2 rows/thread, double-buffered b64 tiles.
//   rank_p[i]    = descending position in predicted order
//                  (== reverse of stable ascending argsort; ties: larger j first)
//   idcg_part[i] = (2^t_i - 1)/log2(pos_true_desc(i)+2)   (tie-order invariant)
// ---------------------------------------------------------------------------
__global__ void lr_rank_kernel(const float2* __restrict__ tp, int n,
                               int* __restrict__ rank_p,
                               float* __restrict__ idcg_part) {
  __shared__ float2 stp[2][TILE];
  const int tid = threadIdx.x;
  const int i0  = blockIdx.x * (2 * TILE) + tid;
  const int i1  = i0 + TILE;
  const float2 a0 = tp[i0];
  const float2 a1 = tp[i1];
  int cp0 = 0, ct0 = 0, cp1 = 0, ct1 = 0;

  lds_cp_f64(&stp[0][tid], &tp[tid]);          // prologue: tile 0 in flight
  int buf = 0;
  for (int base = 0; base < n; base += TILE) {
    const bool has_next = (base + TILE) < n;
    if (has_next) {                            // issue tile m+1 under tile m's compute
      lds_cp_f64(&stp[buf ^ 1][tid], &tp[base + TILE + tid]);
      lds_cp_wait1();                          // tile m resident (m+1 still in flight)
    } else {
      lds_cp_wait0();
    }
    __syncthreads();
#pragma unroll 8
    for (int k = 0; k < TILE; ++k) {
      const int    j = base + k;
      const float2 c = stp[buf][k];
      cp0 += (c.y > a0.y) || ((c.y == a0.y) && (j > i0));
      ct0 += (c.x > a0.x) || ((c.x == a0.x) && (j < i0));
      cp1 += (c.y > a1.y) || ((c.y == a1.y) && (j > i1));
      ct1 += (c.x > a1.x) || ((c.x == a1.x) && (j < i1));
    }
    __syncthreads();                           // all reads of buf done before reuse
    buf ^= 1;
  }
  rank_p[i0]    = cp0;
  rank_p[i1]    = cp1;
  idcg_part[i0] = (exp2f(a0.x) - 1.0f) / __log2f((float)(ct0 + 2));
  idcg_part[i1] = (exp2f(a1.x) - 1.0f) / __log2f((float)(ct1 + 2));
}

// ---------------------------------------------------------------------------
// Kernel 2: build per-position AoS record {t_orig, gain, ps, invd}.
//   pack[i].x = t[i]               (mask side: ORIGINAL array at position)
//   pack[i].w = 1/log2(i+2)
//   pack[r].y = 2^t_i - 1  with r = rank_p[i]   (predicted-desc order)
//   pack[r].z = p[i]
// rank_p is a permutation -> every component written exactly once.
// ---------------------------------------------------------------------------
__global__ void lr_scatter_kernel(const float* __restrict__ t,
                                  const float* __restrict__ p,
                                  const int* __restrict__ rank_p,
                                  float4* __restrict__ pack, int n) {
  const int i = blockIdx.x * blockDim.x + threadIdx.x;
  if (i < n) {
    const int   pos = rank_p[i];
    const float ti  = t[i];
    pack[i].x   = ti;
    pack[i].w   = 1.0f / __log2f((float)(i + 2));
    pack[pos].y = exp2f(ti) - 1.0f;
    pack[pos].z = p[i];
  }
}

// ---------------------------------------------------------------------------
// Kernel 3: deterministic fixed-order reduction of idcg_part -> idcg scalar.
// ---------------------------------------------------------------------------
__global__ void lr_idcg_reduce_kernel(const float* __restrict__ part,
                                      float* __restrict__ out, int n) {
  __shared__ float s[1024];
  float a = 0.0f;
  for (int j = threadIdx.x; j < n; j += 1024) a += part[j];
  s[threadIdx.x] = a;
  __syncthreads();
  for (int off = 512; off > 0; off >>= 1) {
    if (threadIdx.x < (unsigned)off) s[threadIdx.x] += s[threadIdx.x + off];
    __syncthreads();
  }
  if (threadIdx.x == 0) out[0] = s[0];
}

// ---------------------------------------------------------------------------
// Kernel 4: pairwise sweep, 2 rows/thread, double-buffered b128 tiles,
// one ds_load_b128 per column. One exp per ordered pair:
// rho(i,j)=1/(1+e), rho(j,i)=1-rho(i,j), z symmetric.
//   contrib =  z*rho       if t_i>t_j && i<j    (+= val[i,j])
//           = -z*(1-rho)   if t_j>t_i && j<i    (-= val[j,i])
// ---------------------------------------------------------------------------
__global__ void lr_pair_kernel(const float4* __restrict__ pack,
                               const float* __restrict__ idcg,
                               float* __restrict__ partial, int n) {
  __shared__ float4 scol[2][TILE];
  const int tid   = threadIdx.x;
  const int i0    = blockIdx.x * (2 * TILE) + tid;
  const int i1    = i0 + TILE;
  const int chunk = blockIdx.y;
  const int cols  = n / NCHUNK;
  const int cbase = chunk * cols;
  const int cend  = cbase + cols;

  const float4 r0 = pack[i0];
  const float4 r1 = pack[i1];

  float acc0 = 0.0f, acc1 = 0.0f;

  lds_cp_f128(&scol[0][tid], &pack[cbase + tid]);   // prologue: tile 0 in flight
  int buf = 0;
  for (int base = cbase; base < cend; base += TILE) {
    const bool has_next = (base + TILE) < cend;
    if (has_next) {                                 // next tile streams under compute
      lds_cp_f128(&scol[buf ^ 1][tid], &pack[base + TILE + tid]);
      lds_cp_wait1();
    } else {
      lds_cp_wait0();
    }
    __syncthreads();
#pragma unroll 4
    for (int k = 0; k < TILE; ++k) {
      const int    j = base + k;
      const float4 c = scol[buf][k];
      {
        const float e  = __expf(r0.z - c.z);
        const float rr = __builtin_amdgcn_rcpf(1.0f + e);
        const float z  = fabsf((r0.y - c.y) * (c.w - r0.w));
        float v = 0.0f;
        if ((r0.x > c.x) && (i0 < j))      v =  z * rr;
        else if ((c.x > r0.x) && (j < i0)) v = -z * (1.0f - rr);
        acc0 += v;
      }
      {
        const float e  = __expf(r1.z - c.z);
        const float rr = __builtin_amdgcn_rcpf(1.0f + e);
        const float z  = fabsf((r1.y - c.y) * (c.w - r1.w));
        float v = 0.0f;
        if ((r1.x > c.x) && (i1 < j))      v =  z * rr;
        else if ((c.x > r1.x) && (j < i1)) v = -z * (1.0f - rr);
        acc1 += v;
      }
    }
    __syncthreads();
    buf ^= 1;
  }
  const float s = 1.0f / idcg[0];
  partial[i0 * NCHUNK + chunk] = acc0 * s;
  partial[i1 * NCHUNK + chunk] = acc1 * s;
}

// ---------------------------------------------------------------------------
// Kernel 5: fixed-order sum of the 8 column-slice partials per row.
// ---------------------------------------------------------------------------
__global__ void lr_finalize_kernel(const float* __restrict__ partial,
                                   float* __restrict__ lambdas, int n) {
  const int i = blockIdx.x * blockDim.x + threadIdx.x;
  if (i < n) {
    float a = 0.0f;
#pragma unroll
    for (int c = 0; c < NCHUNK; ++c) a += partial[i * NCHUNK + c];
    lambdas[i] = a;
  }
}

// ---------------------------------------------------------------------------
extern "C" void kernel_launch(void* const* d_in, const int* in_sizes, int n_in,
                              void* d_out, int out_size, void* d_ws, size_t ws_size,
                              hipStream_t stream) {
  (void)n_in; (void)out_size; (void)ws_size;
  const float* t = (const float*)d_in[0];   // true_scores  (f32)
  const float* p = (const float*)d_in[1];   // predicted    (f32)
  const int n    = in_sizes[0];             // 8192
  float* lambdas = (float*)d_out;

  // Workspace layout (bytes); every word read is written earlier in this call.
  char*   ws        = (char*)d_ws;
  int*    rank_p    = (int*)   (ws);                           // n ints
  float*  idcg_part = (float*) (ws + (size_t)n * 4);           // n floats
  float2* tp2       = (float2*)(ws + (size_t)n * 8);           // n float2
  float4* pack      = (float4*)(ws + (size_t)n * 16);          // n float4
  float*  idcg      = (float*) (ws + (size_t)n * 32);          // 1 float
  float*  partial   = (float*) (ws + (size_t)n * 32 + 256);    // n*NCHUNK floats

  const dim3 blk(TILE);
  const dim3 grid1((n + TILE - 1) / TILE);
  const dim3 gridR(n / (2 * TILE));           // 2 rows per thread
  const dim3 gridP(n / (2 * TILE), NCHUNK);   // 2 rows per thread x 8 col slices

  lr_pack_kernel<<<grid1, blk, 0, stream>>>(t, p, tp2, n);
  lr_rank_kernel<<<gridR, blk, 0, stream>>>(tp2, n, rank_p, idcg_part);
  lr_scatter_kernel<<<grid1, blk, 0, stream>>>(t, p, rank_p, pack, n);
  lr_idcg_reduce_kernel<<<1, 1024, 0, stream>>>(idcg_part, idcg, n);
  lr_pair_kernel<<<gridP, blk, 0, stream>>>(pack, idcg, partial, n);
  lr_finalize_kernel<<<grid1, blk, 0, stream>>>(partial, lambdas, n);
}